// NodeModel_7713761264052
// MI455X (gfx1250) — compile-verified
//
#include <hip/hip_runtime.h>
#include <hip/hip_bf16.h>

// ---------------------------------------------------------------------------
// GNN NodeModel for MI455X (gfx1250, wave32).
//   aggr[row[e]] += edge_attr[e]                 (scatter-add, BW/atomic bound)
//   out = relu([x‖aggr] @ W1 + b1) @ W2 + b2     (fused MLP, V_WMMA_F32_16X16X4_F32)
// fp32 WMMA keeps reference precision; the whole problem is HBM-bandwidth bound
// (~450 MB scatter traffic -> ~20 us @ 23.3 TB/s dominates; MLP ~77 MB, 4.9 GFLOP).
// Weights are stored in LDS pair-interleaved ([k/2][n][2]) so every WMMA B
// fragment is ONE aligned ds_load_b64 straight into an even VGPR pair.
// ---------------------------------------------------------------------------

typedef float v2f __attribute__((ext_vector_type(2)));
typedef float v8f __attribute__((ext_vector_type(8)));

#define DIN    64
#define DOUT   64
#define TWOD   128   // 2*DIN, rows of W1
#define HIDDEN 128
#define PR1    288   // W1 pair-row pitch: 2*128 + 32  (32 mod 64 banks -> halves disjoint)
#define PR2    160   // W2 pair-row pitch: 2*64  + 32
#define H_LD   132   // h-tile pitch: 128 + 4 -> 16 A rows hit 16 distinct bank groups
#define WAVES_PER_BLOCK 4
#define MLP_BLOCK (WAVES_PER_BLOCK * 32)

__global__ void zero_f4_kernel(float4* __restrict__ p, int n4) {
  int i = blockIdx.x * blockDim.x + threadIdx.x;
  if (i < n4) p[i] = make_float4(0.f, 0.f, 0.f, 0.f);
}

// One thread per float4 chunk of one edge: 4x global_atomic_add_f32.
__global__ void scatter_add_kernel(const float* __restrict__ edge_attr,
                                   const long long* __restrict__ edge_row,
                                   float* __restrict__ aggr,
                                   int n_edges) {
  long long t = (long long)blockIdx.x * blockDim.x + threadIdx.x;
  if (t >= (long long)n_edges * (DIN / 4)) return;
  int e  = (int)(t >> 4);
  int c4 = ((int)t & 15) << 2;
  int row = (int)edge_row[e];   // edge_index[0][e]
  float4 v = *(const float4*)(edge_attr + (size_t)e * DIN + c4);
  float* dst = aggr + (size_t)row * DIN + c4;
  atomicAdd(dst + 0, v.x);
  atomicAdd(dst + 1, v.y);
  atomicAdd(dst + 2, v.z);
  atomicAdd(dst + 3, v.w);
}

// Fused MLP. One wave per 16-node tile; weights + per-wave h-tiles in LDS.
__global__ __launch_bounds__(MLP_BLOCK) void mlp_wmma_kernel(
    const float* __restrict__ x, const float* __restrict__ aggr,
    const float* __restrict__ W1, const float* __restrict__ b1,
    const float* __restrict__ W2, const float* __restrict__ b2,
    float* __restrict__ out, int n_nodes)
{
  extern __shared__ float smem[];
  float* sW1 = smem;                       // 64 pair-rows * PR1
  float* sW2 = sW1 + 64 * PR1;             // 64 pair-rows * PR2
  float* sB1 = sW2 + 64 * PR2;             // [128]
  float* sB2 = sB1 + HIDDEN;               // [64]
  float* sH  = sB2 + DOUT;                 // WAVES_PER_BLOCK * [16][H_LD]

  const int tid  = threadIdx.x;
  const int wave = tid >> 5;
  const int lane = tid & 31;
  const int half = lane >> 4;   // 0: lanes 0-15, 1: lanes 16-31
  const int mn   = lane & 15;   // M (A/C rows) or N (B/C cols)
  const int kk   = half << 1;   // K sub-offset per ISA f32 fragment layout

  // Stage weights pair-interleaved: sW[(k/2)*PR + 2n + (k&1)] = W[k][n].
  for (int i = tid; i < TWOD * HIDDEN; i += MLP_BLOCK) {
    int k = i >> 7, n = i & 127;
    sW1[(k >> 1) * PR1 + (n << 1) + (k & 1)] = W1[i];
  }
  for (int i = tid; i < HIDDEN * DOUT; i += MLP_BLOCK) {
    int k = i >> 6, n = i & 63;
    sW2[(k >> 1) * PR2 + (n << 1) + (k & 1)] = W2[i];
  }
  if (tid < HIDDEN) sB1[tid] = b1[tid];
  if (tid < DOUT)   sB2[tid] = b2[tid];
  __syncthreads();

  const int row0 = (blockIdx.x * WAVES_PER_BLOCK + wave) << 4;
  float* hb = sH + wave * (16 * H_LD);     // wave-private 16x128 tile

  // Load h = [x | aggr] tile: 16 rows x 32 float4-chunks, 32 lanes.
  for (int c = lane; c < 16 * (TWOD / 4); c += 32) {
    int r   = c >> 5;
    int col = (c & 31) << 2;
    int node = row0 + r;
    float4 v = make_float4(0.f, 0.f, 0.f, 0.f);
    if (node < n_nodes)
      v = (col < DIN) ? *(const float4*)(x    + (size_t)node * DIN + col)
                      : *(const float4*)(aggr + (size_t)node * DIN + (col - DIN));
    *(float4*)(hb + r * H_LD + col) = v;
  }
  // Same-wave LDS is in-order; no block barrier needed for wave-private tile.

  // Per-lane fragment base pointers (all fragment fetches become b64 loads
  // with immediate offsets off these).
  const float* hA = hb + mn * H_LD + kk;              // A: + k0
  const float* w1f = sW1 + half * PR1 + (mn << 1);    // B1: + (k0/2)*PR1 + 32n
  const float* w2f = sW2 + half * PR2 + (mn << 1);    // B2: + (k0/2)*PR2 + 32n

  // ---- GEMM1: h16x128 = relu(in16x128 @ W1 + b1), 8 N-tiles of 16 ----
  {
    v8f acc[8] = {};
    for (int k0 = 0; k0 < TWOD; k0 += 4) {
      v2f a = *(const v2f*)(hA + k0);
      const float* wk = w1f + (k0 >> 1) * PR1;
#pragma unroll
      for (int n = 0; n < 8; ++n) {
        v2f bf = *(const v2f*)(wk + (n << 5));
        acc[n] = __builtin_amdgcn_wmma_f32_16x16x4_f32(
            false, a, false, bf, (short)0, acc[n], false, false);
      }
    }
    // bias + ReLU, write back into the same wave-private LDS tile
#pragma unroll
    for (int n = 0; n < 8; ++n) {
      int colc = (n << 4) + mn;
      float bias = sB1[colc];
#pragma unroll
      for (int v = 0; v < 8; ++v) {            // C: M = v + 8*half, N = colc
        float hv = acc[n][v] + bias;
        hb[((half << 3) + v) * H_LD + colc] = fmaxf(hv, 0.f);
      }
    }
  }

  // ---- GEMM2: out16x64 = h @ W2 + b2, 4 N-tiles of 16 ----
  {
    v8f acc[4] = {};
    for (int k0 = 0; k0 < HIDDEN; k0 += 4) {
      v2f a = *(const v2f*)(hA + k0);
      const float* wk = w2f + (k0 >> 1) * PR2;
#pragma unroll
      for (int n = 0; n < 4; ++n) {
        v2f bf = *(const v2f*)(wk + (n << 5));
        acc[n] = __builtin_amdgcn_wmma_f32_16x16x4_f32(
            false, a, false, bf, (short)0, acc[n], false, false);
      }
    }
#pragma unroll
    for (int n = 0; n < 4; ++n) {
      int colc = (n << 4) + mn;
      float bias = sB2[colc];
#pragma unroll
      for (int v = 0; v < 8; ++v) {
        int node = row0 + (half << 3) + v;
        if (node < n_nodes)
          out[(size_t)node * DOUT + colc] = acc[n][v] + bias;  // coalesced
      }
    }
  }
}

extern "C" void kernel_launch(void* const* d_in, const int* in_sizes, int n_in,
                              void* d_out, int out_size, void* d_ws, size_t ws_size,
                              hipStream_t stream) {
  // setup_inputs order: x, edge_index, edge_attr, u, batch, W1, b1, W2, b2
  const float*     xp    = (const float*)d_in[0];
  const long long* eidx  = (const long long*)d_in[1];  // int64 [2, E]; row = first E
  const float*     eattr = (const float*)d_in[2];
  const float*     W1    = (const float*)d_in[5];
  const float*     b1    = (const float*)d_in[6];
  const float*     W2    = (const float*)d_in[7];
  const float*     b2    = (const float*)d_in[8];
  float* outp = (float*)d_out;
  float* aggr = (float*)d_ws;          // [n_nodes, 64] f32 scratch

  const int n_nodes = in_sizes[0] / DIN;
  const int n_edges = in_sizes[2] / DIN;

  // 1) zero aggregation buffer (workspace is poisoned by harness)
  int n4 = n_nodes * (DIN / 4);
  zero_f4_kernel<<<(n4 + 255) / 256, 256, 0, stream>>>((float4*)aggr, n4);

  // 2) scatter-add edge features
  long long sc_threads = (long long)n_edges * (DIN / 4);
  int sc_blocks = (int)((sc_threads + 255) / 256);
  scatter_add_kernel<<<sc_blocks, 256, 0, stream>>>(eattr, eidx, aggr, n_edges);

  // 3) fused MLP with fp32 WMMA
  int mb = (n_nodes + WAVES_PER_BLOCK * 16 - 1) / (WAVES_PER_BLOCK * 16);
  size_t shmem = (size_t)(64 * PR1 + 64 * PR2 + HIDDEN + DOUT +
                          WAVES_PER_BLOCK * 16 * H_LD) * sizeof(float);
  mlp_wmma_kernel<<<mb, MLP_BLOCK, shmem, stream>>>(xp, aggr, W1, b1, W2, b2,
                                                    outp, n_nodes);
}